// FeatureClustering_9062380995393
// MI455X (gfx1250) — compile-verified
//
#include <hip/hip_runtime.h>
#include <hip/hip_bf16.h>
#include <math.h>

typedef float v2f __attribute__((ext_vector_type(2)));
typedef float v4f __attribute__((ext_vector_type(4)));
typedef float v8f __attribute__((ext_vector_type(8)));

#define RTOT 65536
#define EDIM 128
#define KCL  12
#define BVAR 4096

// LDS layout (floats)
#define ROT_OFF   0
#define A_OFF     16384
#define UNIT_OFF  (A_OFF + 16384)
#define TRANS_OFF (UNIT_OFF + 2048)
#define ISTD2_OFF (TRANS_OFF + 128)
#define LSTD_OFF  (ISTD2_OFF + 128)
#define INORM_OFF (LSTD_OFF + 128)
#define S2_OFF    (INORM_OFF + 16)
#define SLS_OFF   (S2_OFF + 128)
#define SMEM_FLOATS (SLS_OFF + 4)

// constants (fp64-evaluated at compile time)
#define LOG2PI_F 1.8378770664093453
#define C0_F  ((float)(-64.0  * LOG2PI_F))   // -(E/2) * log(2*pi)
#define C1_F  ((float)(-63.5  * LOG2PI_F))   // -((E-1)/2) * log(2*pi)
#define SQRT2_F   1.41421356237309515f
#define SQRTPI_F  1.77245385090551603f

__device__ __forceinline__ float logerfc_f(float z) {
    if (z > 5.0f) {
        float z2 = z * z;
        float z4 = z2 * z2;
        float z6 = z4 * z2;
        return -z2 - logf(z * SQRTPI_F)
               + log1pf(-1.0f / (2.0f * z2) + 3.0f / (4.0f * z4) - 15.0f / (8.0f * z6));
    }
    return logf(erfcf(z));
}

// ---------------- Kernel 0: exclusive scan of counts -> offsets ----------------
__global__ void fc_scan_counts(const int* __restrict__ counts, int* __restrict__ offsets, int B) {
    __shared__ int part[256];
    const int tid = threadIdx.x;
    const int per = B / 256;            // 16 for B=4096
    const int base = tid * per;
    int local[16];
    int s = 0;
    for (int i = 0; i < per; ++i) { local[i] = counts[base + i]; s += local[i]; }
    part[tid] = s;
    __syncthreads();
    for (int st = 1; st < 256; st <<= 1) {
        int v = (tid >= st) ? part[tid - st] : 0;
        __syncthreads();
        part[tid] += v;
        __syncthreads();
    }
    int run = part[tid] - s;            // exclusive base for this chunk
    for (int i = 0; i < per; ++i) { offsets[base + i] = run; run += local[i]; }
}

// ---------------- Kernel 1: fused transform + per-read log-likelihoods --------
// grid: R/128 blocks, 256 threads (8 waves, 16 rows per wave)
__global__ void __launch_bounds__(256, 1)
fc_main(const float* __restrict__ alt, const float* __restrict__ trans,
        const float* __restrict__ rot, const float* __restrict__ stdev,
        const float* __restrict__ dirs, const float* __restrict__ mu,
        const float* __restrict__ sigma, const float* __restrict__ lambd,
        const float* __restrict__ astd, float* __restrict__ ll) {
    extern __shared__ float smem[];
    float* rot_s   = smem + ROT_OFF;     // 16384 : rot[n][k] row-major (B-matrix source)
    float* a_s     = smem + A_OFF;       // 16384 : raw alt strip, later x strip
    float* unit_s  = smem + UNIT_OFF;    // 2048  : unit dirs padded to 16 rows
    float* trans_s = smem + TRANS_OFF;   // 128
    float* istd2_s = smem + ISTD2_OFF;   // 128   : 1/sigma_e^2
    float* lstd_s  = smem + LSTD_OFF;    // 128   : log sigma_e
    float* inorm_s = smem + INORM_OFF;   // 16    : 1/||dir_k||
    float* s2_s    = smem + S2_OFF;      // 128   : per-row ||x||^2 (8 waves * 16)
    float* sls_s   = smem + SLS_OFF;     // 4     : sum log sigma_e

    const int tid  = threadIdx.x;
    const int lane = tid & 31;
    const int wv   = tid >> 5;
    const int rowBase = blockIdx.x * 128;
    const int r16  = lane & 15;
    const int koff = (lane >> 4) << 1;   // lanes 0-15 -> K,K+1 ; lanes 16-31 -> K+2,K+3
    const int wbase = wv * 16 * EDIM;

    // ---- phase 1: async-stage rot (64KB) + alt strip (64KB) into LDS ----
    {
        const unsigned long long rbase = (unsigned long long)(uintptr_t)rot;
        const unsigned long long abase =
            (unsigned long long)(uintptr_t)(alt + (size_t)rowBase * EDIM);
#pragma unroll
        for (int i = 0; i < 16; ++i) {
            unsigned goff = (unsigned)(tid + i * 256) * 16u;        // byte offset
            unsigned ldst_rot = (unsigned)(ROT_OFF * 4) + goff;     // LDS byte addr
            unsigned ldst_a   = (unsigned)(A_OFF * 4) + goff;
            asm volatile("global_load_async_to_lds_b128 %0, %1, %2"
                         :: "v"(ldst_rot), "v"(goff), "s"(rbase) : "memory");
            asm volatile("global_load_async_to_lds_b128 %0, %1, %2"
                         :: "v"(ldst_a), "v"(goff), "s"(abase) : "memory");
        }
    }
    // small tensors staged normally while async copies are in flight
    if (tid < 128) {
        trans_s[tid] = trans[tid];
        float s = stdev[tid];
        istd2_s[tid] = 1.0f / (s * s);
        lstd_s[tid] = logf(s);
    }
    if (tid < KCL) {
        float n2 = 0.0f;
        for (int e = 0; e < EDIM; ++e) { float v = dirs[tid * EDIM + e]; n2 += v * v; }
        inorm_s[tid] = rsqrtf(n2);
    }
    asm volatile("s_wait_asynccnt 0x0" ::: "memory");
    __syncthreads();

    // ---- phase 2: unit matrix (needs inv-norms), sum-log-std ----
    for (int i = tid; i < 2048; i += 256) {
        int n = i >> 7, e = i & 127;
        unit_s[i] = (n < KCL) ? dirs[n * EDIM + e] * inorm_s[n] : 0.0f;
    }
    if (tid == 0) {
        float s = 0.0f;
        for (int e = 0; e < EDIM; ++e) s += lstd_s[e];
        sls_s[0] = s;
    }
    __syncthreads();

    // ---- GEMM: x[16,128] = (a + trans)[16,128] @ rot^T  (fp32 WMMA 16x16x4) ----
    v8f zero = {0.f, 0.f, 0.f, 0.f, 0.f, 0.f, 0.f, 0.f};
    v8f acc[8];
#pragma unroll
    for (int i = 0; i < 8; ++i) acc[i] = zero;
#pragma unroll 4
    for (int kc = 0; kc < 32; ++kc) {
        v2f af = *(v2f*)&a_s[wbase + r16 * EDIM + kc * 4 + koff];
        v2f tf = *(v2f*)&trans_s[kc * 4 + koff];
        af.x += tf.x; af.y += tf.y;                 // fold translation into A fragment
#pragma unroll
        for (int nt = 0; nt < 8; ++nt) {
            v2f bf = *(v2f*)&rot_s[(nt * 16 + r16) * EDIM + kc * 4 + koff];
            acc[nt] = __builtin_amdgcn_wmma_f32_16x16x4_f32(
                false, af, false, bf, (short)0, acc[nt], false, false);
        }
    }

    // ---- write x back into this wave's LDS strip (overwrites consumed A) ----
#pragma unroll
    for (int nt = 0; nt < 8; ++nt) {
#pragma unroll
        for (int j = 0; j < 8; ++j) {
            int M = (lane < 16) ? j : (j + 8);
            a_s[wbase + M * EDIM + nt * 16 + r16] = acc[nt][j];
        }
    }

    // ---- skinny GEMM: dot[16,16] = x[16,128] @ unit^T (K padded to 16) ----
    v8f dacc = zero;
#pragma unroll 4
    for (int kc = 0; kc < 32; ++kc) {
        v2f af = *(v2f*)&a_s[wbase + r16 * EDIM + kc * 4 + koff];
        v2f bf = *(v2f*)&unit_s[r16 * EDIM + kc * 4 + koff];
        dacc = __builtin_amdgcn_wmma_f32_16x16x4_f32(
            false, af, false, bf, (short)0, dacc, false, false);
    }

    // ---- per-row s1 = sum (x/std)^2, s2 = sum x^2 (half-row per lane) ----
    float s1 = 0.0f, s2 = 0.0f;
    const int half = (lane >> 4) * 64;
#pragma unroll
    for (int c = 0; c < 16; ++c) {
        v4f x = *(v4f*)&a_s[wbase + r16 * EDIM + half + c * 4];
        v4f w = *(v4f*)&istd2_s[half + c * 4];
        s2 += x.x * x.x + x.y * x.y + x.z * x.z + x.w * x.w;
        s1 += x.x * x.x * w.x + x.y * x.y * w.y + x.z * x.z * w.z + x.w * x.w * w.w;
    }
    s1 += __shfl_xor(s1, 16, 32);
    s2 += __shfl_xor(s2, 16, 32);
    if (lane < 16) {
        float nonart = C0_F - sls_s[0] - 0.5f * s1;
        ll[(size_t)(rowBase + wv * 16 + lane) * 13] = nonart;
        s2_s[wv * 16 + lane] = s2;
    }

    // ---- EMG + orthogonal gaussian per (row, k); dot tile: lane = k, vgpr = row ----
    const int k = r16;
    if (k < KCL) {
        float muk = mu[k], sg = sigma[k], lam = lambd[k], as_ = astd[k];
        float var = sg * sg;
        float inv2as2 = 0.5f / (as_ * as_);
        float lgas = logf(as_);
        float lglam = logf(0.5f * lam);
        float zden = 1.0f / (SQRT2_F * sg);
        float m2 = muk + lam * var;
#pragma unroll
        for (int j = 0; j < 8; ++j) {
            int M = (lane < 16) ? j : (j + 8);
            float dot = dacc[j];
            float s2r = s2_s[wv * 16 + M];
            float orth2 = fmaxf(s2r - dot * dot, 0.0f);
            float orth_ll = C1_F - 127.0f * lgas - orth2 * inv2as2;
            float z = (m2 - dot) * zden;
            float par_ll = lglam + logerfc_f(z) + 0.5f * lam * (m2 + muk - 2.0f * dot);
            ll[(size_t)(rowBase + wv * 16 + M) * 13 + 1 + k] = orth_ll + par_ll;
        }
    }
}

// ---------------- Kernel 2: segment sums + logsumexp + capped logits ----------
// one wave per variant; 8 variants per 256-thread block
__global__ void fc_finalize(const float* __restrict__ ll, const int* __restrict__ offsets,
                            const int* __restrict__ counts, const float* __restrict__ w,
                            float* __restrict__ out, int B) {
    __shared__ float red[8 * 16];
    const int tid = threadIdx.x;
    const int lane = tid & 31;
    const int wv = tid >> 5;
    const int b = blockIdx.x * 8 + wv;
    if (b < B && lane < 13) {
        int off = offsets[b], cnt = counts[b];
        float s = 0.0f;
        for (int i = 0; i < cnt; ++i) s += ll[(size_t)(off + i) * 13 + lane];
        if (lane >= 1) {
            // log_softmax of cluster weights (computed redundantly, K=12)
            float m = -INFINITY;
            for (int kk = 0; kk < KCL; ++kk) m = fmaxf(m, w[kk]);
            float se = 0.0f;
            for (int kk = 0; kk < KCL; ++kk) se += expf(w[kk] - m);
            s += w[lane - 1] - m - logf(se);
        }
        out[(size_t)B + (size_t)b * 13 + lane] = s;
        red[wv * 16 + lane] = s;
    }
    __syncthreads();
    if (b < B && lane == 0) {
        float nonart = red[wv * 16];
        float m = -INFINITY;
        for (int kk = 1; kk < 13; ++kk) m = fmaxf(m, red[wv * 16 + kk]);
        float se = 0.0f;
        for (int kk = 1; kk < 13; ++kk) se += expf(red[wv * 16 + kk] - m);
        float logit = (m + logf(se)) - nonart;
        out[b] = 20.0f * tanhf(logit / 20.0f);
    }
}

extern "C" void kernel_launch(void* const* d_in, const int* in_sizes, int n_in,
                              void* d_out, int out_size, void* d_ws, size_t ws_size,
                              hipStream_t stream) {
    const float* alt    = (const float*)d_in[0];
    // d_in[1] (ref_re) is unused by the reference result
    const float* trans  = (const float*)d_in[2];
    const float* rot    = (const float*)d_in[3];
    const float* stdev  = (const float*)d_in[4];
    const float* dirs   = (const float*)d_in[5];
    const float* mu     = (const float*)d_in[6];
    const float* sigma  = (const float*)d_in[7];
    const float* lambd  = (const float*)d_in[8];
    const float* astd   = (const float*)d_in[9];
    const float* wts    = (const float*)d_in[10];
    const int*   counts = (const int*)d_in[11];
    float* out = (float*)d_out;

    const int R = in_sizes[0] / EDIM;    // 65536
    const int B = in_sizes[11];          // 4096

    int*   offsets = (int*)d_ws;
    float* ll      = (float*)((char*)d_ws + (size_t)B * sizeof(int));

    fc_scan_counts<<<1, 256, 0, stream>>>(counts, offsets, B);

    const size_t shmem = (size_t)SMEM_FLOATS * sizeof(float);
    fc_main<<<R / 128, 256, shmem, stream>>>(alt, trans, rot, stdev, dirs,
                                             mu, sigma, lambd, astd, ll);

    fc_finalize<<<(B + 7) / 8, 256, 0, stream>>>(ll, offsets, counts, wts, out, B);
}